// GaussianRadiusExtractor_73675868996057
// MI455X (gfx1250) — compile-verified
//
#include <hip/hip_runtime.h>
#include <hip/hip_bf16.h>
#include <math.h>

// Problem constants (from the reference): nb=2, nf=2, nx=256, ny=256, nz=16
#define NR   24        // NB_RADIUSES
#define NZ   16
#define NBF  4         // nb*nf
#define NXY  65536     // 256*256 plane points
#define NY   256

constexpr float PI_F        = 3.14159265358979323846f;
constexpr float NEG_INV_2S2 = -50.0f;               // -1/(2*SIGMA^2), SIGMA=0.1
constexpr float TWO_COEF    = 7.9788456080286535f;  // 2 / (SIGMA*sqrt(2*pi))

// Per-radius branch selection and folded targets (computed in double, matches
// numpy: angles = r*(2pi/24) - pi; fold by +/-pi when |angle| > pi/2 strictly).
constexpr bool USE_NEG[NR] = {
  true,  true,  true,  true,  true,  true,                       // r = 0..5
  false, false, false, false, false, false, false,               // r = 6..12
  false, false, false, false, false, false,                      // r = 13..18
  true,  true,  true,  true,  true                               // r = 19..23
};
constexpr float TGT[NR] = {
   0.0f,         0.26179939f,  0.52359878f,  0.78539816f,  1.04719755f,  1.30899694f,
  -1.57079633f, -1.30899694f, -1.04719755f, -0.78539816f, -0.52359878f, -0.26179939f, 0.0f,
   0.26179939f,  0.52359878f,  0.78539816f,  1.04719755f,  1.30899694f,  1.57079633f,
  -1.30899694f, -1.04719755f, -0.78539816f, -0.52359878f, -0.26179939f
};

// ---------------------------------------------------------------------------
// Kernel 1: transcendental-bound weighted reduction over the plane.
// Block = 256 threads = 16 z-lanes x 16 pixel-slots -> every global load of x
// is a fully contiguous 1 KiB line (z is the innermost axis of x).
// Each thread keeps 24 fp32 accumulators; per pixel: 1 atan2f + 24 v_exp_f32.
// ---------------------------------------------------------------------------
__global__ __launch_bounds__(256)
void k1_gauss_partial(const float* __restrict__ x,
                      const float* __restrict__ com_real,
                      const float* __restrict__ com_imag,
                      float* __restrict__ partial,   // [bf][z][r][chunk]
                      int nchunk, int chunk_pix)
{
  const int bf    = blockIdx.y;          // 0..3  (b*nf + f)
  const int chunk = blockIdx.x;          // 0..nchunk-1
  const int tid   = threadIdx.x;
  const int z     = tid & (NZ - 1);      // z lane
  const int slot  = tid >> 4;            // pixel slot 0..15

  const float cr = com_real[bf * NZ + z];   // column center (real axis)
  const float ci = com_imag[bf * NZ + z];   // row center    (imag axis)

  float acc[NR];
  #pragma unroll
  for (int r = 0; r < NR; ++r) acc[r] = 0.0f;

  const float* __restrict__ xb = x + (size_t)bf * ((size_t)NXY * NZ);
  const int p0 = chunk * chunk_pix;

  for (int p = p0 + slot; p < p0 + chunk_pix; p += 16) {
    const int   px  = p >> 8;         // row  (imag axis index)
    const int   py  = p & (NY - 1);   // col  (real axis index)
    const float dxv = (float)py - cr;
    const float dyv = (float)px - ci;
    const float th  = atan2f(dyv, dxv);
    // arg(-v) = th -/+ pi; sign mismatch vs atan2(-dy,-dx) only at |diff|>=pi/2
    // where exp(-50*d^2) underflows to 0 -> numerically identical.
    const float thn = (th > 0.0f) ? (th - PI_F) : (th + PI_F);
    const float xv  = xb[(size_t)p * NZ + z];
    #pragma unroll
    for (int r = 0; r < NR; ++r) {
      const float d = (USE_NEG[r] ? thn : th) - TGT[r];
      acc[r] = fmaf(__expf(d * d * NEG_INV_2S2), xv, acc[r]);   // v_exp_f32 + v_fmac
    }
  }

  // Lanes l and l+16 share the same z (wave32): pairwise reduce the two slots.
  #pragma unroll
  for (int r = 0; r < NR; ++r) acc[r] += __shfl_xor(acc[r], 16, 32);

  __shared__ float red[8 * NZ * NR];   // 12 KiB of the 320 KiB WGP LDS
  const int wave = tid >> 5;
  const int lane = tid & 31;
  if (lane < NZ) {                     // lane == z here
    #pragma unroll
    for (int r = 0; r < NR; ++r) red[(wave * NZ + z) * NR + r] = acc[r];
  }
  __syncthreads();

  // Deterministic cross-wave sum, then emit per-chunk partials.
  for (int j = tid; j < NZ * NR; j += 256) {
    float s = 0.0f;
    #pragma unroll
    for (int w = 0; w < 8; ++w) s += red[w * NZ * NR + j];
    const int zz = j / NR;
    const int r  = j - zz * NR;
    partial[(((size_t)bf * NZ + zz) * NR + r) * nchunk + chunk] = s;
  }
}

// ---------------------------------------------------------------------------
// Kernel 2: reduce partials over chunks with V_WMMA_F32_16X16X4_F32.
// Column-sum-as-GEMM: D = ones(16x4) * B(4x16) + C, B[k][n] = partial[chunk k][radius n].
// Full fp32 matrix op (CDNA5-only shape) -> exact-precision reduction.
// One wave per (b,f,z) combo; two 16-column radius tiles cover NR=24.
// ---------------------------------------------------------------------------
typedef __attribute__((ext_vector_type(2))) float v2f;
typedef __attribute__((ext_vector_type(8))) float v8f;

__device__ __forceinline__ float wmma_colsum(const float* __restrict__ rowPtr,
                                             int nchunk, int khalf, float mask)
{
  v8f c = {0.0f, 0.0f, 0.0f, 0.0f, 0.0f, 0.0f, 0.0f, 0.0f};
  const v2f a = {1.0f, 1.0f};   // ones matrix: layout-invariant
  for (int ks = 0; ks < nchunk; ks += 4) {
    // B 4x16 f32: VGPR0 holds K=khalf*2+0, VGPR1 holds K=khalf*2+1 (K-striped
    // across lane halves, mirroring the documented 16x4 A layout).
    v2f b;
    b.x = rowPtr[ks + khalf * 2 + 0] * mask;
    b.y = rowPtr[ks + khalf * 2 + 1] * mask;
    c = __builtin_amdgcn_wmma_f32_16x16x4_f32(false, a, false, b,
                                              (short)0, c, false, false);
  }
  return c[0];   // every row of D equals the column sum for this lane's n
}

__global__ __launch_bounds__(32)
void k2_wmma_reduce(const float* __restrict__ partial,  // [bf][z][r][chunk]
                    float* __restrict__ out,            // [bf][r][z]
                    int nchunk)
{
  const int combo = blockIdx.x;        // 0..63 = bf*16 + z
  const int lane  = threadIdx.x;       // one wave, EXEC all-ones through WMMA
  const int n     = lane & 15;         // radius column within tile
  const int khalf = lane >> 4;         // which K pair this lane half feeds

  const int bf = combo >> 4;
  const int z  = combo & 15;
  const float* __restrict__ Pc = partial + (size_t)combo * NR * nchunk;

  // Tile 0: radii 0..15
  const float S0 = wmma_colsum(Pc + (size_t)n * nchunk, nchunk, khalf, 1.0f);
  // Tile 1: radii 16..23 (columns 24..31 masked to zero, branchlessly)
  const int   r1  = 16 + n;
  const int   rc1 = (r1 < NR) ? r1 : (NR - 1);
  const float m1  = (r1 < NR) ? 1.0f : 0.0f;
  const float S1 = wmma_colsum(Pc + (size_t)rc1 * nchunk, nchunk, khalf, m1);

  if (lane < 16) {
    out[((size_t)bf * NR + n) * NZ + z] = sqrtf(TWO_COEF * S0);
    if (r1 < NR)
      out[((size_t)bf * NR + r1) * NZ + z] = sqrtf(TWO_COEF * S1);
  }
}

// ---------------------------------------------------------------------------
extern "C" void kernel_launch(void* const* d_in, const int* in_sizes, int n_in,
                              void* d_out, int out_size, void* d_ws, size_t ws_size,
                              hipStream_t stream)
{
  (void)in_sizes; (void)n_in; (void)out_size;
  const float* x        = (const float*)d_in[0];  // (2,2,256,256,16) f32
  const float* com_real = (const float*)d_in[1];  // (2,2,16) f32
  const float* com_imag = (const float*)d_in[2];  // (2,2,16) f32
  float*       out      = (float*)d_out;          // (2,2,24,16) f32
  float*       P        = (float*)d_ws;           // partials scratch

  // 128 plane chunks -> 512 blocks (4096 waves) for WGP occupancy; shrink if
  // the workspace is unexpectedly small (deterministic in ws_size).
  int nchunk = 128;
  while (nchunk > 4 &&
         (size_t)NBF * NZ * NR * (size_t)nchunk * sizeof(float) > ws_size)
    nchunk >>= 1;
  const int chunk_pix = NXY / nchunk;

  dim3 g1(nchunk, NBF);
  k1_gauss_partial<<<g1, 256, 0, stream>>>(x, com_real, com_imag, P,
                                           nchunk, chunk_pix);
  k2_wmma_reduce<<<NBF * NZ, 32, 0, stream>>>(P, out, nchunk);
}